// BandPass_40355512713376
// MI455X (gfx1250) — compile-verified
//
#include <hip/hip_runtime.h>

typedef __attribute__((ext_vector_type(2))) float v2f;
typedef __attribute__((ext_vector_type(4))) float v4f;
typedef __attribute__((ext_vector_type(8))) float v8f;

// Only check builtin availability in the device pass; the host (x86) pass
// does not register AMDGPU builtins for __has_builtin.
#if defined(__HIP_DEVICE_COMPILE__) && defined(__gfx1250__)
#if !__has_builtin(__builtin_amdgcn_wmma_f32_16x16x4_f32)
#error "wmma_f32_16x16x4_f32 builtin not available on this toolchain (device pass)"
#endif
#endif

#define T_LEN   480000
#define CHUNK   800      // samples per chunk (multiple of 16); 600 chunks/row
#define WARM    128      // warm-up samples; pole^128 ~ 3e-21 -> exact in fp32
#define NCHUNK  (T_LEN / CHUNK)

// One wave handles 16 rows (one row-group) over one chunk.
// Columns N = 16 batch rows, M = 16 timesteps, K = 16 inputs + 2 state (+2 pad).
__global__ __launch_bounds__(256) void bandpass_wmma_kernel(
    const float* __restrict__ x, float* __restrict__ y)
{
    const int ln  = threadIdx.x & 31;
    const int wg  = blockIdx.x * 8 + (threadIdx.x >> 5);   // 0..2399
    const int rg  = wg & 3;                                 // row group 0..3
    const int c   = wg >> 2;                                // chunk 0..599
    if (c >= NCHUNK) return;

    const bool lo = (ln < 16);
    const int  m  = ln & 15;                                // column / M row
    const int  row = rg * 16 + m;
    const float* __restrict__ xr = x + (size_t)row * T_LEN;
    float*       __restrict__ yr = y + (size_t)row * T_LEN;

    // ---- coefficients (double, matching the reference's Python math) ----
    const double sinw0 = 0.49999999999999994;   // sin(pi/6)
    const double cosw0 = 0.8660254037844387;    // cos(pi/6)
    const double alpha = sinw0 / (2.0 * 0.707);
    const double a0d   = 1.0 + alpha;
    const double b0d   = alpha / a0d;
    const double a1d   = (-2.0 * cosw0) / a0d;
    const double a2d   = (1.0 - alpha) / a0d;

    // AR impulse response h[0..16]
    double h[17];
    h[0] = 1.0;
    h[1] = -a1d;
    for (int j = 2; j <= 16; ++j) h[j] = -a1d * h[j - 1] - a2d * h[j - 2];

    // G[m][k]: block coefficient matrix (b0 folded into the data columns)
    auto G = [&](int mm, int k) -> double {
        if (k < 16)  return (k <= mm) ? b0d * h[mm - k] : 0.0;
        if (k == 16) return h[mm + 1];          // multiplies y[-1]
        if (k == 17) return -a2d * h[mm];       // multiplies y[-2]
        return 0.0;                             // K padding
    };

    // A fragments: 16x4 f32 layout -> lane n(0-15): (K=4i+0, 4i+1) of row n,
    //              lane n+16: (K=4i+2, 4i+3) of row n.
    v2f afrag[5];
#pragma unroll
    for (int i = 0; i < 5; ++i) {
        const int k0 = 4 * i + (lo ? 0 : 2);
        afrag[i].x = (float)G(m, k0);
        afrag[i].y = (float)G(m, k0 + 1);
    }

    const int tStart = c * CHUNK;
    const int tEnd   = tStart + CHUNK;
    int t = (c == 0) ? 0 : (tStart - WARM);

    v2f stateB; stateB.x = 0.0f; stateB.y = 0.0f;   // (y[-1], y[-2])

    for (; t < tEnd; t += 16) {
        // Load x[t-4 .. t+16) : 5 aligned float4 loads (t is a multiple of 16).
        float buf[20];
        if (t >= 4) {
            const v4f* p = (const v4f*)(xr + (t - 4));
#pragma unroll
            for (int q = 0; q < 5; ++q) {
                v4f v = p[q];
                buf[4 * q + 0] = v.x; buf[4 * q + 1] = v.y;
                buf[4 * q + 2] = v.z; buf[4 * q + 3] = v.w;
            }
        } else {  // only the global first block of each row
#pragma unroll
            for (int j = 0; j < 20; ++j) {
                const int gi = t - 4 + j;
                buf[j] = (gi >= 0) ? xr[gi] : 0.0f;
            }
        }

        // dff[k+2] = x[t+k] - x[t+k-2]  (buf coords: buf[j] = x[t-4+j])
        float dff[18];
#pragma unroll
        for (int j = 2; j <= 17; ++j) dff[j] = buf[j + 2] - buf[j];

        // B fragments: lane n supplies (K=4i,4i+1); lane n+16 (K=4i+2,4i+3)
        v2f b0f, b1f, b2f, b3f, b4f;
        b0f.x = lo ? dff[2]  : dff[4];   b0f.y = lo ? dff[3]  : dff[5];
        b1f.x = lo ? dff[6]  : dff[8];   b1f.y = lo ? dff[7]  : dff[9];
        b2f.x = lo ? dff[10] : dff[12];  b2f.y = lo ? dff[11] : dff[13];
        b3f.x = lo ? dff[14] : dff[16];  b3f.y = lo ? dff[15] : dff[17];
        b4f.x = lo ? stateB.x : 0.0f;    b4f.y = lo ? stateB.y : 0.0f;

        v8f acc = {0.f, 0.f, 0.f, 0.f, 0.f, 0.f, 0.f, 0.f};
        acc = __builtin_amdgcn_wmma_f32_16x16x4_f32(false, afrag[0], false, b0f,
                                                    (short)0, acc, false, false);
        acc = __builtin_amdgcn_wmma_f32_16x16x4_f32(false, afrag[1], false, b1f,
                                                    (short)0, acc, false, false);
        acc = __builtin_amdgcn_wmma_f32_16x16x4_f32(false, afrag[2], false, b2f,
                                                    (short)0, acc, false, false);
        acc = __builtin_amdgcn_wmma_f32_16x16x4_f32(false, afrag[3], false, b3f,
                                                    (short)0, acc, false, false);
        acc = __builtin_amdgcn_wmma_f32_16x16x4_f32(false, afrag[4], false, b4f,
                                                    (short)0, acc, false, false);

        // D layout: vgpr j, lanes 0-15 -> y[t+j]; lanes 16-31 -> y[t+8+j]
        if (t >= tStart) {
            const int ob = t + (lo ? 0 : 8);
            v4f o0; o0.x = acc[0]; o0.y = acc[1]; o0.z = acc[2]; o0.w = acc[3];
            v4f o1; o1.x = acc[4]; o1.y = acc[5]; o1.z = acc[6]; o1.w = acc[7];
            *(v4f*)(yr + ob)     = o0;
            *(v4f*)(yr + ob + 4) = o1;
        }

        // Next block's state: y[15] lives in acc[7] of lane m+16, y[14] in acc[6].
        const float y15 = __shfl(acc[7], m + 16, 32);
        const float y14 = __shfl(acc[6], m + 16, 32);
        stateB.x = y15;
        stateB.y = y14;
    }
}

extern "C" void kernel_launch(void* const* d_in, const int* in_sizes, int n_in,
                              void* d_out, int out_size, void* d_ws, size_t ws_size,
                              hipStream_t stream) {
    (void)in_sizes; (void)n_in; (void)d_ws; (void)ws_size; (void)out_size;
    const float* x = (const float*)d_in[0];
    float* y = (float*)d_out;
    // 2400 waves = 4 row-groups x 600 chunks; 8 waves per 256-thread block.
    const int totalWaves = 4 * NCHUNK;          // 2400
    const int blocks = totalWaves / 8;          // 300
    bandpass_wmma_kernel<<<blocks, 256, 0, stream>>>(x, y);
}